// QSAModule_27161373180359
// MI455X (gfx1250) — compile-verified
//
#include <hip/hip_runtime.h>
#include <hip/hip_bf16.h>
#include <math.h>

typedef __attribute__((ext_vector_type(2))) float v2f;
typedef __attribute__((ext_vector_type(8))) float v8f;

#define NQ 12
#define NL 4

__device__ __forceinline__ float2 cmul2(float2 a, float2 b) {
    return make_float2(a.x * b.x - a.y * b.y, a.x * b.y + a.y * b.x);
}
__device__ __forceinline__ float2 cadd2(float2 a, float2 b) {
    return make_float2(a.x + b.x, a.y + b.y);
}

// ---------------------------------------------------------------------------
// Kernel 1: per-head, per-qubit 2x2 complex unitary
//   U = prod_l Rz(t2/2) Ry(t1/2) Rz(t0/2)
// Output: Uws[(head*12+q)*8] = {u00r,u00i,u01r,u01i,u10r,u10i,u11r,u11i}
// ---------------------------------------------------------------------------
__global__ void vqc_unitaries_kernel(const float* __restrict__ thQ,
                                     const float* __restrict__ thK,
                                     const float* __restrict__ thV,
                                     float* __restrict__ Uws) {
    int tid = threadIdx.x;
    if (tid >= 3 * NQ) return;
    int head = tid / NQ;
    int q    = tid % NQ;
    const float* th = (head == 0) ? thQ : (head == 1) ? thK : thV;

    float2 u00 = {1.f, 0.f}, u01 = {0.f, 0.f};
    float2 u10 = {0.f, 0.f}, u11 = {1.f, 0.f};

    for (int l = 0; l < NL; ++l) {
        float p0 = 0.5f * th[(l * NQ + q) * 3 + 0];
        float p1 = 0.5f * th[(l * NQ + q) * 3 + 1];
        float p2 = 0.5f * th[(l * NQ + q) * 3 + 2];
        float c0 = cosf(p0), s0 = sinf(p0);
        float c1 = cosf(p1), s1 = sinf(p1);
        float c2 = cosf(p2), s2 = sinf(p2);
        // M = Ry(p1) @ Rz(p0) ; Rz diag(e^{-ip0}, e^{+ip0})
        float2 m00 = { c1 * c0, -c1 * s0};
        float2 m01 = {-s1 * c0, -s1 * s0};
        float2 m10 = { s1 * c0, -s1 * s0};
        float2 m11 = { c1 * c0,  c1 * s0};
        // Ul = Rz(p2) @ M : row0 *= e^{-ip2}, row1 *= e^{+ip2}
        float2 f  = {c2, -s2};
        float2 fc = {c2,  s2};
        float2 a00 = cmul2(f,  m00), a01 = cmul2(f,  m01);
        float2 a10 = cmul2(fc, m10), a11 = cmul2(fc, m11);
        // U = Ul @ U
        float2 n00 = cadd2(cmul2(a00, u00), cmul2(a01, u10));
        float2 n01 = cadd2(cmul2(a00, u01), cmul2(a01, u11));
        float2 n10 = cadd2(cmul2(a10, u00), cmul2(a11, u10));
        float2 n11 = cadd2(cmul2(a10, u01), cmul2(a11, u11));
        u00 = n00; u01 = n01; u10 = n10; u11 = n11;
    }

    float* o = Uws + (head * NQ + q) * 8;
    o[0] = u00.x; o[1] = u00.y;
    o[2] = u01.x; o[3] = u01.y;
    o[4] = u10.x; o[5] = u10.y;
    o[6] = u11.x; o[7] = u11.y;
}

// ---------------------------------------------------------------------------
// Kernel 2: one wave per (head, b, s) task.
//   v'[q] = U[q] @ (cos a/2, sin a/2)            (12 complex 2-vectors)
//   A[khi] = prod_{q=0..5}  v'[q][bit], khi bit5 = qubit0   (64 complex)
//   B[klo] = prod_{q=6..11} v'[q][bit], klo bit5 = qubit6   (64 complex)
//   psi[khi*64+klo] = A_re*B_re - A_im*B_im
// Final expansion as 64x2 @ 2x64 GEMM via 16x V_WMMA_F32_16X16X4_F32.
// ---------------------------------------------------------------------------
__global__ __launch_bounds__(128) void vqc_state_kernel(
    const float* __restrict__ angles,   // (B*S, NQ)
    const float* __restrict__ Uws,      // (3, NQ, 8)
    float* __restrict__ out,            // (3, B*S, 4096) concat
    int bs_total)
{
    __shared__ float2 shV[4][2 * NQ];   // [wave][qubit*2 + component]
    __shared__ float2 shA[4][64];
    __shared__ float2 shB[4][64];

    const int w    = threadIdx.x >> 5;
    const int lane = threadIdx.x & 31;
    const int task = blockIdx.x * 4 + w;          // grid sized exactly
    const int head = task / bs_total;
    const int bs   = task - head * bs_total;

    // --- per-qubit rotated state vectors -------------------------------
    if (lane < NQ) {
        const float* U = Uws + (head * NQ + lane) * 8;
        float u00r = U[0], u00i = U[1], u01r = U[2], u01i = U[3];
        float u10r = U[4], u10i = U[5], u11r = U[6], u11i = U[7];
        float a  = angles[bs * NQ + lane];
        float ca = cosf(0.5f * a);
        float sa = sinf(0.5f * a);
        shV[w][lane * 2 + 0] = make_float2(u00r * ca + u01r * sa,
                                           u00i * ca + u01i * sa);
        shV[w][lane * 2 + 1] = make_float2(u10r * ca + u11r * sa,
                                           u10i * ca + u11i * sa);
    }
    __syncthreads();

    // --- half Kronecker products (each lane does entries i and i+32) ---
    #pragma unroll
    for (int half = 0; half < 2; ++half) {
        int i = lane + half * 32;
        float2 accA = shV[w][0 * 2 + ((i >> 5) & 1)];
        #pragma unroll
        for (int q = 1; q < 6; ++q)
            accA = cmul2(accA, shV[w][q * 2 + ((i >> (5 - q)) & 1)]);
        shA[w][i] = accA;

        float2 accB = shV[w][6 * 2 + ((i >> 5) & 1)];
        #pragma unroll
        for (int q = 7; q < 12; ++q)
            accB = cmul2(accB, shV[w][q * 2 + ((i >> (11 - q)) & 1)]);
        shB[w][i] = accB;
    }
    __syncthreads();

    // --- 64x64 output = [A_re A_im] @ [B_re ; -B_im] via WMMA ----------
    float* outp = out + (size_t)task * 4096;
    const float msk    = (lane < 16) ? 1.f : 0.f;  // zero K=2,3 lanes
    const int   ll     = lane & 15;
    const int   rowadd = (lane >> 4) << 3;         // +8 rows for upper half

    #pragma unroll
    for (int mt = 0; mt < 4; ++mt) {
        float2 av = shA[w][mt * 16 + ll];
        v2f Am;
        Am[0] = av.x * msk;       // K=0 : A_re
        Am[1] = av.y * msk;       // K=1 : A_im
        #pragma unroll
        for (int nt = 0; nt < 4; ++nt) {
            float2 bv = shB[w][nt * 16 + ll];
            v2f Bm;
            Bm[0] =  bv.x * msk;  // K=0 row : B_re
            Bm[1] = -bv.y * msk;  // K=1 row : -B_im
            v8f c = {};
            v8f d = __builtin_amdgcn_wmma_f32_16x16x4_f32(
                false, Am, false, Bm, (short)0, c, false, false);
            float* tp = outp + (mt * 16 + rowadd) * 64 + nt * 16 + ll;
            #pragma unroll
            for (int r = 0; r < 8; ++r)
                tp[r * 64] = d[r];
        }
    }
}

// ---------------------------------------------------------------------------
extern "C" void kernel_launch(void* const* d_in, const int* in_sizes, int n_in,
                              void* d_out, int out_size, void* d_ws, size_t ws_size,
                              hipStream_t stream) {
    const float* angles = (const float*)d_in[0];
    const float* thQ    = (const float*)d_in[1];
    const float* thK    = (const float*)d_in[2];
    const float* thV    = (const float*)d_in[3];
    float* Uws = (float*)d_ws;                 // 3*12*8 floats = 1152 B

    const int bs_total = in_sizes[0] / NQ;     // B*S = 4096
    const int tasks    = 3 * bs_total;         // 12288, divisible by 4

    vqc_unitaries_kernel<<<1, 64, 0, stream>>>(thQ, thK, thV, Uws);
    vqc_state_kernel<<<tasks / 4, 128, 0, stream>>>(
        angles, Uws, (float*)d_out, bs_total);
}